// ArchetipesNetwork_40793599377658
// MI455X (gfx1250) — compile-verified
//
#include <hip/hip_runtime.h>
#include <math.h>

// Problem constants (from reference): T=256, I=128, M=64, H=512, DT=0.01
#define T_  256
#define I_  128
#define M_  64
#define H_  512
#define DT_ 0.01f

typedef float v2f __attribute__((ext_vector_type(2)));
typedef float v8f __attribute__((ext_vector_type(8)));

// ---------------------------------------------------------------------------
// init: states_all[0] = initial_states ; seq_fb[0] = 0 (outs0 == 0 -> fb==0)
// ---------------------------------------------------------------------------
__global__ __launch_bounds__(256) void init_kernel(
    const float* __restrict__ init_states,
    float* __restrict__ states_all,
    float* __restrict__ seq_fb)
{
    int i = blockIdx.x * blockDim.x + threadIdx.x;
    if (i < M_ * 2 * H_) states_all[i] = init_states[i];
    if (i < M_ * H_)     seq_fb[i]     = 0.0f;
}

// ---------------------------------------------------------------------------
// FB GEMM 1:  G(64x512) = hy_prev(64x512) @ Wm^T(512x512)
//   hy_prev row m lives at states_all[t][m][0][:], row stride 2*H.
//   B[k][n] = Wm[n][k]  -> B fragment is a float2 load from row n of Wm.
//   WMMA f32 16x16x4; one wave owns a 16(M) x 64(N) strip (A reused 4x).
//   Fragment layout (ISA 7.12.2, 32-bit 16x4 A / 4x16 B):
//     lanes 0-15 hold K = k0+0,k0+1 ; lanes 16-31 hold K = k0+2,k0+3
//   D layout: lane l, vgpr v -> M = v + 8*(l>=16), N = l&15.
// ---------------------------------------------------------------------------
__global__ __launch_bounds__(256) void fb_gemm1_kernel(
    const float* __restrict__ states_all,
    const float* __restrict__ Wm,
    float* __restrict__ G, int t)
{
    const int lane  = threadIdx.x & 31;
    const int wave  = threadIdx.x >> 5;
    const int mt    = blockIdx.x;          // 0..3  (M tile)
    const int n0    = wave * 64;           // N strip base (0..448)
    const int l15   = lane & 15;
    const int khalf = (lane >> 4) << 1;    // 0 or 2

    const float* hy    = states_all + (size_t)t * (M_ * 2 * H_);
    const float* arow  = hy + (size_t)(mt * 16 + l15) * (2 * H_);   // hy row, stride 2H
    const float* brow0 = Wm + (size_t)(n0 + l15) * H_;              // Wm row n0+l15

    v8f acc0 = {}, acc1 = {}, acc2 = {}, acc3 = {};
    for (int k0 = 0; k0 < H_; k0 += 4) {
        v2f a  = *(const v2f*)(arow + k0 + khalf);
        v2f b0 = *(const v2f*)(brow0 + (size_t)(0 * 16) * H_ + k0 + khalf);
        v2f b1 = *(const v2f*)(brow0 + (size_t)(1 * 16) * H_ + k0 + khalf);
        v2f b2 = *(const v2f*)(brow0 + (size_t)(2 * 16) * H_ + k0 + khalf);
        v2f b3 = *(const v2f*)(brow0 + (size_t)(3 * 16) * H_ + k0 + khalf);
        acc0 = __builtin_amdgcn_wmma_f32_16x16x4_f32(false, a, false, b0, (short)0, acc0, false, false);
        acc1 = __builtin_amdgcn_wmma_f32_16x16x4_f32(false, a, false, b1, (short)0, acc1, false, false);
        acc2 = __builtin_amdgcn_wmma_f32_16x16x4_f32(false, a, false, b2, (short)0, acc2, false, false);
        acc3 = __builtin_amdgcn_wmma_f32_16x16x4_f32(false, a, false, b3, (short)0, acc3, false, false);
    }

    const int hi = (lane >> 4) * 8;
    float* gout = G + (size_t)(mt * 16 + hi) * H_ + n0 + l15;
#pragma unroll
    for (int v = 0; v < 8; ++v) {
        gout[(size_t)v * H_ +  0] = acc0[v];
        gout[(size_t)v * H_ + 16] = acc1[v];
        gout[(size_t)v * H_ + 32] = acc2[v];
        gout[(size_t)v * H_ + 48] = acc3[v];
    }
}

// ---------------------------------------------------------------------------
// FB GEMM 2:  fb_t(64x512) = C(64x64) @ G(64x512), written to seq_fb[t].
//   B is G directly (no transpose): b.x = G[k][n], b.y = G[k+1][n].
// ---------------------------------------------------------------------------
__global__ __launch_bounds__(256) void fb_gemm2_kernel(
    const float* __restrict__ Cmat,
    const float* __restrict__ G,
    float* __restrict__ fb_t)
{
    const int lane  = threadIdx.x & 31;
    const int wave  = threadIdx.x >> 5;
    const int mt    = blockIdx.x;          // 0..3
    const int n0    = wave * 64;
    const int l15   = lane & 15;
    const int khalf = (lane >> 4) << 1;

    const float* arow = Cmat + (size_t)(mt * 16 + l15) * M_;
    const float* bcol = G + n0 + l15;

    v8f acc0 = {}, acc1 = {}, acc2 = {}, acc3 = {};
    for (int k0 = 0; k0 < M_; k0 += 4) {
        v2f a = *(const v2f*)(arow + k0 + khalf);
        const int kb = k0 + khalf;
        v2f b0, b1, b2, b3;
        b0.x = bcol[(size_t)kb * H_ +  0]; b0.y = bcol[(size_t)(kb + 1) * H_ +  0];
        b1.x = bcol[(size_t)kb * H_ + 16]; b1.y = bcol[(size_t)(kb + 1) * H_ + 16];
        b2.x = bcol[(size_t)kb * H_ + 32]; b2.y = bcol[(size_t)(kb + 1) * H_ + 32];
        b3.x = bcol[(size_t)kb * H_ + 48]; b3.y = bcol[(size_t)(kb + 1) * H_ + 48];
        acc0 = __builtin_amdgcn_wmma_f32_16x16x4_f32(false, a, false, b0, (short)0, acc0, false, false);
        acc1 = __builtin_amdgcn_wmma_f32_16x16x4_f32(false, a, false, b1, (short)0, acc1, false, false);
        acc2 = __builtin_amdgcn_wmma_f32_16x16x4_f32(false, a, false, b2, (short)0, acc2, false, false);
        acc3 = __builtin_amdgcn_wmma_f32_16x16x4_f32(false, a, false, b3, (short)0, acc3, false, false);
    }

    const int hi = (lane >> 4) * 8;
    float* out = fb_t + (size_t)(mt * 16 + hi) * H_ + n0 + l15;
#pragma unroll
    for (int v = 0; v < 8; ++v) {
        out[(size_t)v * H_ +  0] = acc0[v];
        out[(size_t)v * H_ + 16] = acc1[v];
        out[(size_t)v * H_ + 32] = acc2[v];
        out[(size_t)v * H_ + 48] = acc3[v];
    }
}

// ---------------------------------------------------------------------------
// Cell update: one block per module m (64 blocks, 256 threads = 8 wave32).
//   pre[h] = dot(Wx[m,h,:], x_t) + dot(Wh[m,h,:], hy[m,:]) + bias + fb
//   This is the L2-bandwidth-bound part (67MB of Wh per step, L2-resident).
//   Wave-per-row dot: lane reads float4 (b128), hy/x_t staged in LDS,
//   wave32 xor-shuffle reduction.
// ---------------------------------------------------------------------------
__global__ __launch_bounds__(256) void cell_kernel(
    const float* __restrict__ x,
    const float* __restrict__ Wx,
    const float* __restrict__ Wh,
    const float* __restrict__ bias,
    const float* __restrict__ gamma,
    const float* __restrict__ eps,
    const float* __restrict__ seq_fb,
    float* __restrict__ states_all,
    int t)
{
    __shared__ float xs[I_];
    __shared__ float hys[H_];

    const int m   = blockIdx.x;
    const int tid = threadIdx.x;

    const float* St = states_all + (size_t)t * (M_ * 2 * H_) + (size_t)m * (2 * H_);
    if (tid < I_) xs[tid] = x[(size_t)t * I_ + tid];
    hys[tid]       = St[tid];          // hy[m][0..255]
    hys[tid + 256] = St[tid + 256];    // hy[m][256..511]
    __syncthreads();

    const int wave = tid >> 5;
    const int lane = tid & 31;
    const float4* hys4 = (const float4*)hys;
    const float4* xs4  = (const float4*)xs;
    const float4  xv   = xs4[lane];    // x_t cols 4*lane .. 4*lane+3

    for (int r = 0; r < 64; ++r) {
        const int h = wave * 64 + r;
        const float4* whr = (const float4*)(Wh + ((size_t)m * H_ + h) * H_);
        float acc = 0.0f;
#pragma unroll
        for (int j = 0; j < 4; ++j) {
            float4 w  = whr[j * 32 + lane];      // global_load_b128, coalesced
            float4 hv = hys4[j * 32 + lane];     // ds_load_b128
            acc += w.x * hv.x + w.y * hv.y + w.z * hv.z + w.w * hv.w;
        }
        const float4* wxr = (const float4*)(Wx + ((size_t)m * H_ + h) * I_);
        float4 wv = wxr[lane];
        acc += wv.x * xv.x + wv.y * xv.y + wv.z * xv.z + wv.w * xv.w;

#pragma unroll
        for (int off = 16; off >= 1; off >>= 1)
            acc += __shfl_xor(acc, off, 32);

        if (lane == 0) {
            const size_t idx = (size_t)m * H_ + h;
            const float pre  = acc + bias[idx] + seq_fb[(size_t)t * (M_ * H_) + idx];
            const float hyv  = hys[h];
            const float hzv  = St[H_ + h];
            const float hz_n = hzv + DT_ * (tanhf(pre) - gamma[idx] * hyv - eps[idx] * hzv);
            const float hy_n = hyv + DT_ * hz_n;
            float* Sn = states_all + (size_t)(t + 1) * (M_ * 2 * H_) + (size_t)m * (2 * H_);
            Sn[h]      = hy_n;
            Sn[H_ + h] = hz_n;
        }
    }
}

// ---------------------------------------------------------------------------
extern "C" void kernel_launch(void* const* d_in, const int* in_sizes, int n_in,
                              void* d_out, int out_size, void* d_ws, size_t ws_size,
                              hipStream_t stream)
{
    (void)in_sizes; (void)n_in; (void)out_size; (void)ws_size;
    const float* x           = (const float*)d_in[0];  // (T, I)
    const float* init_states = (const float*)d_in[1];  // (M, 2, H)
    const float* Wm          = (const float*)d_in[2];  // (H, H)
    const float* Cmat        = (const float*)d_in[3];  // (M, M)
    const float* Wx          = (const float*)d_in[4];  // (M, H, I)
    const float* Wh          = (const float*)d_in[5];  // (M, H, H)
    const float* bias        = (const float*)d_in[6];  // (M, H)
    const float* gamma       = (const float*)d_in[7];  // (M, H)
    const float* eps         = (const float*)d_in[8];  // (M, H)

    float* states_all = (float*)d_out;                          // (T+1, M, 2, H)
    float* seq_fb     = states_all + (size_t)(T_ + 1) * M_ * 2 * H_;  // (T, M, H)
    float* G          = (float*)d_ws;                           // (M, H) scratch, 128 KB

    init_kernel<<<(M_ * 2 * H_ + 255) / 256, 256, 0, stream>>>(init_states, states_all, seq_fb);

    for (int t = 0; t < T_; ++t) {
        if (t > 0) {
            fb_gemm1_kernel<<<4, 256, 0, stream>>>(states_all, Wm, G, t);
            fb_gemm2_kernel<<<4, 256, 0, stream>>>(Cmat, G, seq_fb + (size_t)t * M_ * H_);
        }
        cell_kernel<<<M_, 256, 0, stream>>>(x, Wx, Wh, bias, gamma, eps,
                                            seq_fb, states_all, t);
    }
}